// SignatureBackend_24575802867997
// MI455X (gfx1250) — compile-verified
//
#include <hip/hip_runtime.h>
#include <math.h>

typedef float v2f __attribute__((ext_vector_type(2)));
typedef float v8f __attribute__((ext_vector_type(8)));

#define WINDOW 60
#define DPATH  9
#define C_IN   8
#define SEQ    2048
#define BATCH  8
#define NWIN   (BATCH * SEQ)      // 16384
#define SIGCH  285
#define KPAD   288                // 285 padded to multiple of 4/16
#define H2     512
#define HID    256
#define ALDS_STRIDE 292           // 288 + 4 pad  (292 % 64 = 36 -> conflict free)
#define HLDS_STRIDE 516           // 512 + 4 pad  (516 % 64 = 4  -> conflict free)

// workspace layout (bytes)
#define LUT_OFF   0
#define W1P_OFF   4096
#define FEAT_OFF  (4096 + KPAD * H2 * 4)   // = 593920

// ---------------------------------------------------------------------------
// Build the Lyndon-word (length-3) index LUT exactly matching the reference's
// nested-loop generation order. 729 entries: output slot or -1.
// ---------------------------------------------------------------------------
__global__ void init_lut3_kernel(int* __restrict__ lut) {
  if (threadIdx.x == 0 && blockIdx.x == 0) {
    int cnt = 0;
    for (int i = 0; i < DPATH; ++i)
      for (int j = 0; j < DPATH; ++j)
        for (int k = 0; k < DPATH; ++k) {
          // (i,j,k) < (j,k,i)
          bool c1 = (i < j) || (i == j && (j < k || (j == k && k < i)));
          // (i,j,k) < (k,i,j)
          bool c2 = (i < k) || (i == k && (j < i || (j == i && k < j)));
          lut[(i * DPATH + j) * DPATH + k] = (c1 && c2) ? cnt++ : -1;
        }
  }
}

// ---------------------------------------------------------------------------
// Zero-pad w1 (285 x 512) -> w1p (288 x 512)
// ---------------------------------------------------------------------------
__global__ void pad_w1_kernel(const float* __restrict__ w1, float* __restrict__ w1p) {
  int idx = blockIdx.x * 256 + threadIdx.x;
  if (idx >= KPAD * H2) return;
  int k = idx / H2, n = idx % H2;
  w1p[idx] = (k < SIGCH) ? w1[k * H2 + n] : 0.0f;
}

// ---------------------------------------------------------------------------
// Log-signature features: one wave32 per window position.
// 81 (i,j) pairs of S2/S3 striped 3-per-lane; per-step increment vector v is
// held one-channel-per-lane and distributed with wave shuffles.
// ---------------------------------------------------------------------------
__global__ __launch_bounds__(256) void logsig_kernel(
    const float* __restrict__ x, const int* __restrict__ lut3,
    float* __restrict__ feats)
{
  __shared__ float xwin[8][WINDOW * C_IN];   // 15360 B
  __shared__ float s2buf[8][81];
  __shared__ float s1buf[8][16];

  const int lane = threadIdx.x & 31;
  const int wv   = threadIdx.x >> 5;
  const int n    = blockIdx.x * 8 + wv;
  const int bI   = n >> 11;            // n / SEQ
  const int t    = n & (SEQ - 1);

  // stage window into LDS (32B per step, float4 x2)
  for (int w = lane; w < WINDOW; w += 32) {
    int p = t + w - (WINDOW - 1);
    if (p < 0) p = 0;
    const float4* src = reinterpret_cast<const float4*>(x + ((size_t)bI * SEQ + p) * C_IN);
    float4* dst = reinterpret_cast<float4*>(&xwin[wv][w * C_IN]);
    dst[0] = src[0];
    dst[1] = src[1];
  }
  __syncthreads();

  // pair assignment: lane owns pairs p0=lane, p1=lane+32, p2=lane+64 (if <81)
  const int p0 = lane;
  const int p1 = lane + 32;
  const int p2r = lane + 64;
  const bool has2 = (p2r < 81);
  const int p2 = has2 ? p2r : 80;
  const int i0 = p0 / 9, j0 = p0 % 9;
  const int i1 = p1 / 9, j1 = p1 % 9;
  const int i2 = p2 / 9, j2 = p2 % 9;

  const int myCh = (lane >= 1 && lane <= 8) ? (lane - 1) : 0;

  float s1Own = 0.0f;                 // S1 component for channel 'lane' (lanes 0..8)
  float S2q0 = 0.0f, S2q1 = 0.0f, S2q2 = 0.0f;
  float S3q0[9], S3q1[9], S3q2[9];
#pragma unroll
  for (int k = 0; k < 9; ++k) { S3q0[k] = 0.f; S3q1[k] = 0.f; S3q2[k] = 0.f; }

  const float dt = 1.0f / (float)(WINDOW - 1);

  for (int w = 0; w < WINDOW; ++w) {
    // per-lane increment component (lane 0 = time channel, lanes 1..8 = data)
    float vOwn;
    if (w == 0) {
      // inc[0] = aug[0] (diff prepend zeros): time=0, data=x[pos0]
      vOwn = (lane == 0) ? 0.0f : xwin[wv][myCh];
    } else {
      vOwn = (lane == 0) ? dt
           : (xwin[wv][w * C_IN + myCh] - xwin[wv][(w - 1) * C_IN + myCh]);
    }

    // replicate v[k] for static k
    float vk[9];
#pragma unroll
    for (int k = 0; k < 9; ++k) vk[k] = __shfl(vOwn, k, 32);

    // per-pair scalars (pre-update S1)
    float vi0 = __shfl(vOwn, i0, 32), vj0 = __shfl(vOwn, j0, 32);
    float vi1 = __shfl(vOwn, i1, 32), vj1 = __shfl(vOwn, j1, 32);
    float vi2 = __shfl(vOwn, i2, 32), vj2 = __shfl(vOwn, j2, 32);
    float s1i0 = __shfl(s1Own, i0, 32);
    float s1i1 = __shfl(s1Own, i1, 32);
    float s1i2 = __shfl(s1Own, i2, 32);

    // nS3[ijk] = S3 + (S2old + 0.5*S1old[i]*v[j] + v[i]*v[j]/6) * v[k]
    float a0 = S2q0 + 0.5f * s1i0 * vj0 + (1.0f / 6.0f) * vi0 * vj0;
    float a1 = S2q1 + 0.5f * s1i1 * vj1 + (1.0f / 6.0f) * vi1 * vj1;
    float a2 = S2q2 + 0.5f * s1i2 * vj2 + (1.0f / 6.0f) * vi2 * vj2;
#pragma unroll
    for (int k = 0; k < 9; ++k) {
      S3q0[k] = fmaf(a0, vk[k], S3q0[k]);
      S3q1[k] = fmaf(a1, vk[k], S3q1[k]);
      S3q2[k] = fmaf(a2, vk[k], S3q2[k]);
    }
    // nS2[ij] = S2 + (S1old[i] + 0.5*v[i]) * v[j]
    S2q0 = fmaf(s1i0 + 0.5f * vi0, vj0, S2q0);
    S2q1 = fmaf(s1i1 + 0.5f * vi1, vj1, S2q1);
    S2q2 = fmaf(s1i2 + 0.5f * vi2, vj2, S2q2);
    // S1 update last
    s1Own += vOwn;
  }

  // mirror S1 / S2 to LDS for dynamic-index reads in extraction
  s2buf[wv][p0] = S2q0;
  s2buf[wv][p1] = S2q1;
  if (has2) s2buf[wv][p2] = S2q2;
  if (lane < 9) s1buf[wv][lane] = s1Own;
  __syncthreads();

  float* f = feats + (size_t)n * KPAD;
  const float* s1 = s1buf[wv];
  const float* s2 = s2buf[wv];

  if (lane < 9) f[lane] = s1Own;                       // L1
  if (lane >= 9 && lane < 12) f[SIGCH + (lane - 9)] = 0.0f;  // K padding zeros

#define EMIT_PAIR(ii, jj, S2V, S3A)                                          \
  {                                                                          \
    const float s1i = s1[ii], s1j = s1[jj];                                  \
    if ((ii) < (jj))                                                         \
      f[9 + (ii) * 8 - (ii) * ((ii) - 1) / 2 + ((jj) - (ii) - 1)] =          \
          (S2V) - 0.5f * s1i * s1j;                                          \
    _Pragma("unroll")                                                        \
    for (int k = 0; k < 9; ++k) {                                            \
      float l3 = S3A[k] - 0.5f * (s1i * s2[(jj) * 9 + k] + (S2V) * s1[k])    \
               + (1.0f / 3.0f) * s1i * s1j * s1[k];                          \
      int o = lut3[((ii) * 9 + (jj)) * 9 + k];                               \
      if (o >= 0) f[45 + o] = l3;                                            \
    }                                                                        \
  }

  EMIT_PAIR(i0, j0, S2q0, S3q0)
  EMIT_PAIR(i1, j1, S2q1, S3q1)
  if (has2) EMIT_PAIR(i2, j2, S2q2, S3q2)
#undef EMIT_PAIR
}

// ---------------------------------------------------------------------------
// Fused MLP: GEMM1 (WMMA f32) + bias + exact GELU -> LDS -> LayerNorm ->
// GEMM2 (WMMA f32) + bias -> out.  16 rows per block, 8 waves of 32.
// ---------------------------------------------------------------------------
__global__ __launch_bounds__(256) void mlp_kernel(
    const float* __restrict__ feats, const float* __restrict__ w1p,
    const float* __restrict__ b1, const float* __restrict__ ln_g,
    const float* __restrict__ ln_b, const float* __restrict__ w2,
    const float* __restrict__ b2, float* __restrict__ out)
{
  __shared__ float a_lds[16 * ALDS_STRIDE];   // 18688 B
  __shared__ float h_lds[16 * HLDS_STRIDE];   // 33024 B

  const int tid  = threadIdx.x;
  const int lane = tid & 31;
  const int wv   = tid >> 5;
  const int m0   = blockIdx.x * 16;

  // stage 16x288 A tile (coalesced)
  for (int idx = tid; idx < 16 * KPAD; idx += 256) {
    int r = idx / KPAD, c = idx - r * KPAD;
    a_lds[r * ALDS_STRIDE + c] = feats[(size_t)(m0 + r) * KPAD + c];
  }
  __syncthreads();

  const int row = lane & 15;           // A-fragment M index
  const int kk2 = (lane >> 4) * 2;     // A/B fragment K sub-offset
  const int col = lane & 15;           // B/C fragment N index
  const int mhi = (lane >> 4) << 3;    // C fragment M offset (0 or 8)

  const v8f zero8 = {0.f, 0.f, 0.f, 0.f, 0.f, 0.f, 0.f, 0.f};

  // ---- GEMM1: 16 x 512, K = 288, each wave covers 64 cols (4 tiles)
  v8f acc[4] = {zero8, zero8, zero8, zero8};
  const int nb1 = wv * 64;
  for (int k0 = 0; k0 < KPAD; k0 += 4) {
    v2f a;
    a.x = a_lds[row * ALDS_STRIDE + k0 + kk2];
    a.y = a_lds[row * ALDS_STRIDE + k0 + kk2 + 1];
#pragma unroll
    for (int tt = 0; tt < 4; ++tt) {
      int ncol = nb1 + tt * 16 + col;
      v2f bb;
      bb.x = w1p[(size_t)(k0 + kk2) * H2 + ncol];
      bb.y = w1p[(size_t)(k0 + kk2 + 1) * H2 + ncol];
      acc[tt] = __builtin_amdgcn_wmma_f32_16x16x4_f32(
          false, a, false, bb, (short)0, acc[tt], false, false);
    }
  }

  // bias + exact GELU -> h_lds
#pragma unroll
  for (int tt = 0; tt < 4; ++tt) {
    int ncol = nb1 + tt * 16 + col;
    float bias = b1[ncol];
#pragma unroll
    for (int e = 0; e < 8; ++e) {
      int M = e + mhi;
      float hv = acc[tt][e] + bias;
      float g = 0.5f * hv * (1.0f + erff(hv * 0.70710678118654752f));
      h_lds[M * HLDS_STRIDE + ncol] = g;
    }
  }
  __syncthreads();

  // ---- LayerNorm over 512, two rows per wave
#pragma unroll
  for (int rr = 0; rr < 2; ++rr) {
    int r = wv * 2 + rr;
    float s = 0.f, ss = 0.f;
#pragma unroll
    for (int i = 0; i < 16; ++i) {
      float v = h_lds[r * HLDS_STRIDE + lane + 32 * i];
      s += v; ss += v * v;
    }
#pragma unroll
    for (int off = 16; off >= 1; off >>= 1) {
      s  += __shfl_xor(s,  off, 32);
      ss += __shfl_xor(ss, off, 32);
    }
    float mean = s * (1.0f / 512.0f);
    float var  = ss * (1.0f / 512.0f) - mean * mean;
    float rstd = rsqrtf(var + 1e-5f);
#pragma unroll
    for (int i = 0; i < 16; ++i) {
      int c = lane + 32 * i;
      float v = h_lds[r * HLDS_STRIDE + c];
      h_lds[r * HLDS_STRIDE + c] = (v - mean) * rstd * ln_g[c] + ln_b[c];
    }
  }
  __syncthreads();

  // ---- GEMM2: 16 x 256, K = 512, each wave covers 32 cols (2 tiles)
  v8f acc2[2] = {zero8, zero8};
  const int nb2 = wv * 32;
  for (int k0 = 0; k0 < H2; k0 += 4) {
    v2f a;
    a.x = h_lds[row * HLDS_STRIDE + k0 + kk2];
    a.y = h_lds[row * HLDS_STRIDE + k0 + kk2 + 1];
#pragma unroll
    for (int tt = 0; tt < 2; ++tt) {
      int ncol = nb2 + tt * 16 + col;
      v2f bb;
      bb.x = w2[(size_t)(k0 + kk2) * HID + ncol];
      bb.y = w2[(size_t)(k0 + kk2 + 1) * HID + ncol];
      acc2[tt] = __builtin_amdgcn_wmma_f32_16x16x4_f32(
          false, a, false, bb, (short)0, acc2[tt], false, false);
    }
  }

#pragma unroll
  for (int tt = 0; tt < 2; ++tt) {
    int ncol = nb2 + tt * 16 + col;
    float bias = b2[ncol];
#pragma unroll
    for (int e = 0; e < 8; ++e) {
      int M = e + mhi;
      out[(size_t)(m0 + M) * HID + ncol] = acc2[tt][e] + bias;
    }
  }
}

// ---------------------------------------------------------------------------
extern "C" void kernel_launch(void* const* d_in, const int* in_sizes, int n_in,
                              void* d_out, int out_size, void* d_ws, size_t ws_size,
                              hipStream_t stream) {
  const float* x    = (const float*)d_in[0];
  const float* w1   = (const float*)d_in[1];
  const float* b1   = (const float*)d_in[2];
  const float* ln_g = (const float*)d_in[3];
  const float* ln_b = (const float*)d_in[4];
  const float* w2   = (const float*)d_in[5];
  const float* b2   = (const float*)d_in[6];
  float* out = (float*)d_out;

  char* ws = (char*)d_ws;
  int*   lut3  = (int*)(ws + LUT_OFF);
  float* w1p   = (float*)(ws + W1P_OFF);
  float* feats = (float*)(ws + FEAT_OFF);

  init_lut3_kernel<<<1, 32, 0, stream>>>(lut3);
  pad_w1_kernel<<<(KPAD * H2 + 255) / 256, 256, 0, stream>>>(w1, w1p);
  logsig_kernel<<<NWIN / 8, 256, 0, stream>>>(x, lut3, feats);
  mlp_kernel<<<NWIN / 16, 256, 0, stream>>>(feats, w1p, b1, ln_g, ln_b, w2, b2, out);
}